// AttDistance_42649025249395
// MI455X (gfx1250) — compile-verified
//
#include <hip/hip_runtime.h>
#include <hip/hip_bf16.h>
#include <stdint.h>

// Problem constants (from reference setup_inputs)
#define BB   4
#define LQ   512
#define LYY  512
#define DDIM 128
// Tiling
#define TQ   64
#define TY   64
#define QSTR 132   // LDS row stride (dwords); 132 mod 64 == 4 -> conflict-free b128 reads
#define YSTR 132

__global__ __launch_bounds__(256)
void att_init_sim(float* __restrict__ sim, int n) {
    int i = blockIdx.x * blockDim.x + threadIdx.x;
    if (i < n) sim[i] = __uint_as_float(0xFF800000u);  // -inf
}

__global__ __launch_bounds__(256)
void att_l1_kernel(const float* __restrict__ q,
                   const float* __restrict__ y,
                   float* __restrict__ att,
                   float* __restrict__ sim) {
    __shared__ float Qs[TQ * QSTR];
    __shared__ float Ys[TY * YSTR];

    const int tid = threadIdx.x;
    const int b  = blockIdx.z;
    const int q0 = blockIdx.y * TQ;
    const int y0 = blockIdx.x * TY;

    const float* __restrict__ qg = q + ((size_t)b * LQ  + q0) * DDIM;
    const float* __restrict__ yg = y + ((size_t)b * LYY + y0) * DDIM;

    // Low 32 bits of a flat shared-memory address are the LDS byte offset
    // (ISA aperture rule).
    const unsigned qs_base = (unsigned)(uintptr_t)&Qs[0];
    const unsigned ys_base = (unsigned)(uintptr_t)&Ys[0];

    // Async global -> LDS copy of both 64x128 f32 tiles (CDNA5 async engine).
    // 2048 float4 slots per tile, 256 threads -> 8 b128 async loads each.
#pragma unroll
    for (int k = 0; k < 8; ++k) {
        const int idx = tid + k * 256;
        const int row = idx >> 5;        // 0..63
        const int c4  = idx & 31;        // float4 column 0..31

        const unsigned lq = qs_base + (unsigned)(row * QSTR + c4 * 4) * 4u;
        const float*   gq = qg + row * DDIM + c4 * 4;
        asm volatile("global_load_async_to_lds_b128 %0, %1, off"
                     :: "v"(lq), "v"(gq) : "memory");

        const unsigned ly = ys_base + (unsigned)(row * YSTR + c4 * 4) * 4u;
        const float*   gy = yg + row * DDIM + c4 * 4;
        asm volatile("global_load_async_to_lds_b128 %0, %1, off"
                     :: "v"(ly), "v"(gy) : "memory");
    }
    // Each wave waits for its own async copies, then workgroup barrier
    // publishes the LDS tiles to all waves.
    asm volatile("s_wait_asynccnt 0" ::: "memory");
    __syncthreads();

    const int tx = tid & 15;   // y-direction lane (16)
    const int ty = tid >> 4;   // q-direction lane (16)

    float acc[4][4];
#pragma unroll
    for (int i = 0; i < 4; ++i)
#pragma unroll
        for (int j = 0; j < 4; ++j) acc[i][j] = 0.0f;

    // 32 float4 chunks over D; unroll x2 for more independent
    // ds_load_b128 -> VALU chains in flight.
#pragma unroll 2
    for (int d = 0; d < DDIM; d += 4) {
        float4 qv[4], yv[4];
#pragma unroll
        for (int i = 0; i < 4; ++i)
            qv[i] = *(const float4*)&Qs[(ty + 16 * i) * QSTR + d];
#pragma unroll
        for (int j = 0; j < 4; ++j)
            yv[j] = *(const float4*)&Ys[(tx + 16 * j) * YSTR + d];
#pragma unroll
        for (int i = 0; i < 4; ++i)
#pragma unroll
            for (int j = 0; j < 4; ++j) {
                acc[i][j] += fabsf(qv[i].x - yv[j].x)
                           + fabsf(qv[i].y - yv[j].y)
                           + fabsf(qv[i].z - yv[j].z)
                           + fabsf(qv[i].w - yv[j].w);
            }
    }

    // Write att = -mean(|diff|) and reduce per-q max into sim.
    float* __restrict__ attb = att + ((size_t)b * LQ + q0) * LYY + y0;
    const float scale = -1.0f / (float)DDIM;

#pragma unroll
    for (int i = 0; i < 4; ++i) {
        const int qr = ty + 16 * i;
        float v0 = acc[i][0] * scale;               // all values <= -0.0f
        attb[(size_t)qr * LYY + tx] = v0;
        float m = v0;
#pragma unroll
        for (int j = 1; j < 4; ++j) {
            const float v = acc[i][j] * scale;
            attb[(size_t)qr * LYY + tx + 16 * j] = v;
            m = fmaxf(m, v);
        }
        // Max across the 16 tx lanes (xor masks < 16 stay inside the group).
#pragma unroll
        for (int off = 1; off < 16; off <<= 1)
            m = fmaxf(m, __shfl_xor(m, off, 32));
        if (tx == 0) {
            // Candidates are all <= -0.0f: float-max == unsigned-min on bits.
            atomicMin((unsigned int*)&sim[(size_t)b * LQ + q0 + qr],
                      __float_as_uint(m));
        }
    }
}

extern "C" void kernel_launch(void* const* d_in, const int* in_sizes, int n_in,
                              void* d_out, int out_size, void* d_ws, size_t ws_size,
                              hipStream_t stream) {
    const float* q = (const float*)d_in[0];
    const float* y = (const float*)d_in[1];

    float* att = (float*)d_out;                       // (B, Lq, Ly)
    float* sim = att + (size_t)BB * LQ * LYY;         // (B, 1, Lq)

    const int nsim = BB * LQ;
    att_init_sim<<<dim3((nsim + 255) / 256), 256, 0, stream>>>(sim, nsim);

    dim3 grid(LYY / TY, LQ / TQ, BB);
    att_l1_kernel<<<grid, dim3(256), 0, stream>>>(q, y, att, sim);
}